// NGCF_61160334295658
// MI455X (gfx1250) — compile-verified
//
#include <hip/hip_runtime.h>
#include <hip/hip_bf16.h>

// ---------------------------------------------------------------------------
// NGCF forward on MI455X (gfx1250, wave32, WMMA).
//
// Layer math (fused):  E' = l2norm(leaky( (S1 + E + S2) @ W + 2b ))
//   S1 = spmm(val,row,col, E),  S2 = spmm(val,row,col, E*E)
// Dense transform: V_WMMA_F32_16X16X4_F32 (fp32-exact matrix pipe).
//   - A fragments preloaded in bulk (batched global_load_b64, no per-step
//     loadcnt-0 stalls)
//   - B staged in LDS in WMMA pair layout -> one aligned ds_load_b64 per
//     fragment, bank-conflict-free between half-waves (160-dword row stride)
// ---------------------------------------------------------------------------

#define NUM_USER 40000
#define NUM_ITEM 60000
#define NN (NUM_USER + NUM_ITEM)   // 100000
#define D 64
#define LLAYERS 3
#define BATCH 1024
#define EMB_STRIDE (D * (LLAYERS + 1))   // 256: [N,256] concat buffer
#define WP_STRIDE 160                    // dwords per k-pair row of W in LDS

typedef __attribute__((ext_vector_type(2))) float v2f;
typedef __attribute__((ext_vector_type(8))) float v8f;

// ---- copy initial embeddings into column block 0 of all_emb [N,256] --------
__global__ void ngcf_init_kernel(const float* __restrict__ uemb,
                                 const float* __restrict__ iemb,
                                 float* __restrict__ all_emb) {
    int t = blockIdx.x * blockDim.x + threadIdx.x;     // one float4 each
    if (t >= NN * (D / 4)) return;
    int r  = t >> 4;          // row
    int c4 = t & 15;          // float4 within the 64-col slice
    float4 v = (r < NUM_USER)
        ? ((const float4*)uemb)[(size_t)r * (D / 4) + c4]
        : ((const float4*)iemb)[(size_t)(r - NUM_USER) * (D / 4) + c4];
    ((float4*)all_emb)[(size_t)r * (EMB_STRIDE / 4) + c4] = v;
}

// ---- zero scratch ----------------------------------------------------------
__global__ void ngcf_zero_kernel(float4* __restrict__ p, long n4) {
    long i = (long)blockIdx.x * blockDim.x + threadIdx.x;
    if (i < n4) p[i] = make_float4(0.f, 0.f, 0.f, 0.f);
}

// ---- SpMM: edge-parallel gather + atomic scatter (L2-resident) -------------
// X points at a 64-wide column slice of all_emb (row stride EMB_STRIDE).
// Each 64-thread group handles one edge; one fp32 lane per feature.
__global__ void ngcf_spmm_kernel(const float* __restrict__ vals,
                                 const int*  __restrict__ rows,
                                 const int*  __restrict__ cols,
                                 const float* __restrict__ X,
                                 float* __restrict__ S1,
                                 float* __restrict__ S2,
                                 int nnz) {
    int e = blockIdx.x * (blockDim.x >> 6) + (threadIdx.x >> 6);
    if (e >= nnz) return;
    int d = threadIdx.x & 63;
    int r = rows[e];
    int c = cols[e];
    float v = vals[e];
    float x = X[(size_t)c * EMB_STRIDE + d];
    atomicAdd(&S1[(size_t)r * D + d], v * x);        // global_atomic_add_f32
    atomicAdd(&S2[(size_t)r * D + d], v * x * x);
}

// ---- fused layer: (S1+E+S2)@W + 2b -> leaky -> l2norm, via WMMA f32 --------
// Block = 128 threads = 4 waves; each wave owns a 16-row tile (64 rows/block).
// A frag (16x4 f32): lane&15 = row, lane>>4 selects K pair {k,k+1} vs {k+2,k+3}.
// C frag (16x16 f32): VGPR r = row r (lanes 0-15) / row r+8 (lanes 16-31).
__global__ __launch_bounds__(128)
void ngcf_layer_kernel(const float* __restrict__ S1,
                       const float* __restrict__ S2,
                       const float* __restrict__ prev,   // stride EMB_STRIDE
                       const float* __restrict__ W,      // [64,64] row-major
                       const float* __restrict__ bias,   // [64]
                       float* __restrict__ out) {        // stride EMB_STRIDE
    // W in "pair layout": Wp[p*WP_STRIDE + 2n + j] = W[2p + j][n], j in {0,1}.
    // Each B fragment = one aligned ds_load_b64. 160-dword row stride makes
    // even-p rows hit banks 0..31 and odd-p rows banks 32..63 for a given
    // column tile -> the two half-waves never collide.
    __shared__ float Wp[32 * WP_STRIDE];    // 20 KB
    __shared__ float bl[D];

    const int tid = threadIdx.x;
    // stage W: each item = (pair-row pr, float4-col c4); 512 items / 128 thr
    #pragma unroll
    for (int it = 0; it < 4; ++it) {
        int item = tid + 128 * it;
        int pr = item >> 4;          // 0..31  (rows 2pr, 2pr+1)
        int c4 = item & 15;          // 0..15  (cols 4c4..4c4+3)
        float4 lo = ((const float4*)W)[(size_t)(2 * pr)     * (D / 4) + c4];
        float4 hi = ((const float4*)W)[(size_t)(2 * pr + 1) * (D / 4) + c4];
        float4* dst = (float4*)&Wp[pr * WP_STRIDE + 8 * c4];
        dst[0] = make_float4(lo.x, hi.x, lo.y, hi.y);   // n = 4c4, 4c4+1
        dst[1] = make_float4(lo.z, hi.z, lo.w, hi.w);   // n = 4c4+2, 4c4+3
    }
    if (tid < D / 4)
        ((float4*)bl)[tid] = ((const float4*)bias)[tid];
    __syncthreads();

    const int lane   = tid & 31;
    const int wave   = tid >> 5;
    const int col    = lane & 15;     // A row select / B-C column select
    const int khalf  = lane >> 4;     // which K pair of the 4-wide step
    const int m_base = blockIdx.x * 64 + wave * 16;

    const int    m  = m_base + col;
    const size_t mc = (size_t)((m < NN) ? m : (NN - 1));   // clamp (EXEC all-1)

    // lane's K offsets are k = 4*kk + 2*khalf -> float2 at (row + 4kk + 2khalf)
    const float* s1p = S1   + mc * D          + 2 * khalf;
    const float* s2p = S2   + mc * D          + 2 * khalf;
    const float* pvp = prev + mc * EMB_STRIDE + 2 * khalf;

    // ---- bulk-preload A fragments: batched independent loads ----
    v2f msum[16], t2[16];
    #pragma unroll
    for (int kk = 0; kk < 16; ++kk) msum[kk] = *(const v2f*)(s1p + 4 * kk);
    #pragma unroll
    for (int kk = 0; kk < 16; ++kk) t2[kk]   = *(const v2f*)(s2p + 4 * kk);
    #pragma unroll
    for (int kk = 0; kk < 16; ++kk) msum[kk] += t2[kk];
    #pragma unroll
    for (int kk = 0; kk < 16; ++kk) t2[kk]   = *(const v2f*)(pvp + 4 * kk);
    #pragma unroll
    for (int kk = 0; kk < 16; ++kk) msum[kk] += t2[kk];

    v8f acc[4] = {v8f{}, v8f{}, v8f{}, v8f{}};

    #pragma unroll
    for (int kk = 0; kk < 16; ++kk) {
        const int p = 2 * kk + khalf;                 // k-pair index
        const float* wrow = &Wp[p * WP_STRIDE];
        #pragma unroll
        for (int t = 0; t < 4; ++t) {
            v2f bf = *(const v2f*)(wrow + 2 * (t * 16 + col));
            acc[t] = __builtin_amdgcn_wmma_f32_16x16x4_f32(
                false, msum[kk], false, bf, (short)0, acc[t], false, false);
        }
    }

    // epilogue: +2b, leaky ReLU(0.2), row L2 norm (rows split by half-wave)
    float ss[8] = {0.f, 0.f, 0.f, 0.f, 0.f, 0.f, 0.f, 0.f};
    #pragma unroll
    for (int t = 0; t < 4; ++t) {
        const float b2 = 2.0f * bl[t * 16 + col];
        #pragma unroll
        for (int r = 0; r < 8; ++r) {
            float y = acc[t][r] + b2;
            y = (y > 0.0f) ? y : 0.2f * y;
            acc[t][r] = y;
            ss[r] += y * y;
        }
    }
    // reduce across the 16 lanes of each half-wave (width-16 keeps halves apart)
    #pragma unroll
    for (int off = 1; off < 16; off <<= 1) {
        #pragma unroll
        for (int r = 0; r < 8; ++r)
            ss[r] += __shfl_xor(ss[r], off, 16);
    }
    float scale[8];
    #pragma unroll
    for (int r = 0; r < 8; ++r)
        scale[r] = 1.0f / fmaxf(sqrtf(ss[r]), 1e-12f);

    #pragma unroll
    for (int r = 0; r < 8; ++r) {
        const int row = m_base + r + 8 * khalf;
        if (row < NN) {
            #pragma unroll
            for (int t = 0; t < 4; ++t)
                out[(size_t)row * EMB_STRIDE + t * 16 + col] = acc[t][r] * scale[r];
        }
    }
}

// ---- final gather of user / pos / neg rows ---------------------------------
__global__ void ngcf_gather_kernel(const int* __restrict__ users,
                                   const int* __restrict__ pos,
                                   const int* __restrict__ neg,
                                   const float* __restrict__ all_emb,
                                   float* __restrict__ out) {
    int j  = blockIdx.x;         // 0 .. 3*BATCH-1
    int d4 = threadIdx.x;        // 0 .. 63 (float4 within 256 cols)
    int src;
    if (j < BATCH)           src = users[j];
    else if (j < 2 * BATCH)  src = NUM_USER + pos[j - BATCH];
    else                     src = NUM_USER + neg[j - 2 * BATCH];
    ((float4*)out)[(size_t)j * (EMB_STRIDE / 4) + d4] =
        ((const float4*)all_emb)[(size_t)src * (EMB_STRIDE / 4) + d4];
}

// ---------------------------------------------------------------------------
extern "C" void kernel_launch(void* const* d_in, const int* in_sizes, int n_in,
                              void* d_out, int out_size, void* d_ws, size_t ws_size,
                              hipStream_t stream) {
    const int*   users = (const int*)d_in[0];
    const int*   pos   = (const int*)d_in[1];
    const int*   neg   = (const int*)d_in[2];
    const int*   erow  = (const int*)d_in[3];
    const int*   ecol  = (const int*)d_in[4];
    const float* evals = (const float*)d_in[5];
    const float* uemb  = (const float*)d_in[6];
    const float* iemb  = (const float*)d_in[7];
    const float* Ws    = (const float*)d_in[8];   // [3,64,64]
    const float* bs    = (const float*)d_in[9];   // [3,64]
    float* out = (float*)d_out;
    const int nnz = in_sizes[3];

    // workspace layout (fp32): all_emb [N,256] | S1 [N,64] | S2 [N,64]
    float* all_emb = (float*)d_ws;
    float* S1 = all_emb + (size_t)NN * EMB_STRIDE;
    float* S2 = S1      + (size_t)NN * D;

    // 1) seed layer-0 embeddings
    {
        int n = NN * (D / 4);
        ngcf_init_kernel<<<(n + 255) / 256, 256, 0, stream>>>(uemb, iemb, all_emb);
    }

    // 2) three propagation layers
    for (int li = 0; li < LLAYERS; ++li) {
        const long n4 = (long)2 * NN * D / 4;     // zero S1 and S2 together
        ngcf_zero_kernel<<<(unsigned)((n4 + 255) / 256), 256, 0, stream>>>(
            (float4*)S1, n4);

        const float* X = all_emb + (size_t)li * D;   // current layer slice
        ngcf_spmm_kernel<<<(nnz + 3) / 4, 256, 0, stream>>>(
            evals, erow, ecol, X, S1, S2, nnz);

        ngcf_layer_kernel<<<(NN + 63) / 64, 128, 0, stream>>>(
            S1, S2, X,
            Ws + (size_t)li * D * D,
            bs + (size_t)li * D,
            all_emb + (size_t)(li + 1) * D);
    }

    // 3) gather (user_e | pos_e | neg_e) -> d_out [3*1024, 256]
    ngcf_gather_kernel<<<3 * BATCH, EMB_STRIDE / 4, 0, stream>>>(
        users, pos, neg, all_emb, out);
}